// HGCNConv_4346506903713
// MI455X (gfx1250) — compile-verified
//
#include <hip/hip_runtime.h>
#include <math.h>

typedef float v2f __attribute__((ext_vector_type(2)));
typedef float v8f __attribute__((ext_vector_type(8)));

#define DD 128
#define MAXNORM 0.996f      /* (1 - 4e-3)/sqrt(c), c = 1 */
#define MIN_NORM 1e-15f
#define ART_CLIP (1.0f - 1e-7f)

__device__ __forceinline__ float artanh_clip(float x) {
    x = fminf(fmaxf(x, -ART_CLIP), ART_CLIP);
    return 0.5f * logf((1.0f + x) / (1.0f - x));
}

__device__ __forceinline__ float red16(float v) {
    v += __shfl_xor(v, 1, 32);
    v += __shfl_xor(v, 2, 32);
    v += __shfl_xor(v, 4, 32);
    v += __shfl_xor(v, 8, 32);
    return v;
}

__device__ __forceinline__ float red32(float v) {
    v = red16(v);
    v += __shfl_xor(v, 16, 32);
    return v;
}

// ---------------- K0: hyp_bias = proj(expmap0(bias)), plus ||hb||^2 ----------------
__global__ void hb_kernel(const float* __restrict__ bias, float* __restrict__ hb) {
    const int l = threadIdx.x;  // 32 lanes, 4 elems each
    float b[4];
    float s = 0.0f;
#pragma unroll
    for (int i = 0; i < 4; ++i) { b[i] = bias[l * 4 + i]; s += b[i] * b[i]; }
    s = red32(s);
    float n  = fmaxf(sqrtf(s), MIN_NORM);
    float f  = tanhf(n) / n;                 // expmap0 factor
    float hn = fmaxf(f * sqrtf(s), MIN_NORM);
    if (hn > MAXNORM) f *= MAXNORM / hn;     // proj
#pragma unroll
    for (int i = 0; i < 4; ++i) hb[l * 4 + i] = b[i] * f;
    if (l == 0) hb[DD] = f * f * s;          // ||hb||^2
}

// ---------------- degree / normalization ----------------
__global__ void zero_u32(unsigned* __restrict__ p, int n) {
    int i = blockIdx.x * blockDim.x + threadIdx.x;
    if (i < n) p[i] = 0u;
}

__global__ void deg_count(const long long* __restrict__ rows, unsigned* __restrict__ deg, int e) {
    int i = blockIdx.x * blockDim.x + threadIdx.x;
    if (i < e) atomicAdd(&deg[(int)rows[i]], 1u);
}

__global__ void dis_kernel(const unsigned* __restrict__ deg, float* __restrict__ dis, int n) {
    int i = blockIdx.x * blockDim.x + threadIdx.x;
    if (i < n) dis[i] = rsqrtf((float)(deg[i] + 1u));  // +1 self loop
}

// ---------------- K2: WMMA GEMM + fused HypLinear + logmap0 -> x_t ----------------
__launch_bounds__(256)
__global__ void gemm_hyp(const float* __restrict__ X, const float* __restrict__ W,
                         const float* __restrict__ HB, float* __restrict__ XT, int nNodes) {
    __shared__ float Wsh[128 * 68];  // half of K at a time, stride 68 -> conflict-free b64 reads

    const int lane = threadIdx.x & 31;
    const int wave = threadIdx.x >> 5;
    const int hi   = lane >> 4;     // K-pair half
    const int lo   = lane & 15;     // row (A) / col (B) within 16
    const int m0   = (blockIdx.x * 8 + wave) * 16;

    v8f acc[8];
#pragma unroll
    for (int t = 0; t < 8; ++t)
#pragma unroll
        for (int j = 0; j < 8; ++j) acc[t][j] = 0.0f;

    const int rowc = min(m0 + lo, nNodes - 1);
    const float* xrow = X + (size_t)rowc * DD;

    float hbv[8];
#pragma unroll
    for (int t = 0; t < 8; ++t) hbv[t] = HB[t * 16 + lo];
    const float hb2 = HB[DD];

    float px = 0.0f;  // partial ||x_row||^2 (this lane's K subset)

    for (int half = 0; half < 2; ++half) {
        __syncthreads();
        {   // cooperative load of W[:, half*64 .. +63] into LDS
            const int r  = threadIdx.x >> 1;
            const int c0 = (threadIdx.x & 1) * 32;
            const float* wsrc = W + (size_t)r * DD + half * 64 + c0;
            float* wdst = Wsh + r * 68 + c0;
#pragma unroll
            for (int i = 0; i < 8; ++i)
                ((float4*)wdst)[i] = ((const float4*)wsrc)[i];
        }
        __syncthreads();

        for (int kk = 0; kk < 64; kk += 4) {
            const int k = half * 64 + kk + 2 * hi;
            float2 av = *(const float2*)(xrow + k);
            v2f a; a.x = av.x; a.y = av.y;
            px += av.x * av.x + av.y * av.y;
#pragma unroll
            for (int t = 0; t < 8; ++t) {
                const float* bp = Wsh + (t * 16 + lo) * 68 + kk + 2 * hi;  // B[k][n] = W[n][k]
                float2 bv = *(const float2*)bp;
                v2f b; b.x = bv.x; b.y = bv.y;
                acc[t] = __builtin_amdgcn_wmma_f32_16x16x4_f32(
                    false, a, false, b, (short)0, acc[t], false, false);
            }
        }
    }

    // full ||x_row||^2 for row `lo` now in every lane
    float xn2full = px + __shfl_xor(px, 16, 32);

    float P[8], Q[8];
#pragma unroll
    for (int j = 0; j < 8; ++j) {
        // per-row reductions (row m0 + j + 8*hi lives in this half's lanes)
        float mn2 = 0.0f, xyv = 0.0f;
#pragma unroll
        for (int t = 0; t < 8; ++t) { float v = acc[t][j]; mn2 += v * v; xyv += v * hbv[t]; }
        mn2 = red16(mn2);
        xyv = red16(xyv);
        float xn2r = __shfl(xn2full, j + 8 * hi, 32);

        float xn = fmaxf(sqrtf(xn2r), MIN_NORM);
        float mn = fmaxf(sqrtf(mn2), MIN_NORM);
        // mobius_matvec scale: res = s * mx
        float s = tanhf(mn / xn * artanh_clip(xn)) / mn;
        // proj(res)
        float rn = fmaxf(s * mn, MIN_NORM);
        if (rn > MAXNORM) { s *= MAXNORM / rn; rn = MAXNORM; }
        float x2 = rn * rn;
        float xy = s * xyv;
        // mobius_add(res, hb)
        float denom = fmaxf(1.0f + 2.0f * xy + x2 * hb2, MIN_NORM);
        float alpha = (1.0f + 2.0f * xy + hb2) / denom;
        float beta  = (1.0f - x2) / denom;
        // ||h||^2 from scalars; proj + logmap0
        float hn2 = alpha * alpha * x2 + 2.0f * alpha * beta * xy + beta * beta * hb2;
        float hn  = fmaxf(sqrtf(fmaxf(hn2, 0.0f)), MIN_NORM);
        float gp  = 1.0f;
        if (hn > MAXNORM) { gp = MAXNORM / hn; hn = MAXNORM; }
        float g = artanh_clip(hn) / hn;
        P[j] = g * gp * alpha * s;
        Q[j] = g * gp * beta;
    }

#pragma unroll
    for (int j = 0; j < 8; ++j) {
        const int m = m0 + j + 8 * hi;
        if (m < nNodes) {
            float* orow = XT + (size_t)m * DD;
#pragma unroll
            for (int t = 0; t < 8; ++t)
                orow[t * 16 + lo] = P[j] * acc[t][j] + Q[j] * hbv[t];
        }
    }
}

// ---------------- K3a: agg = dis[i]^2 * x_t[i]   (self loop term, also inits d_out) ----
__global__ void agg_init(const float* __restrict__ xt, const float* __restrict__ dis,
                         float* __restrict__ out, int n) {
    int i = blockIdx.x * blockDim.x + threadIdx.x;  // i indexes float4s, 32 per node
    if (i < n * 32) {
        int node = i >> 5;
        float w = dis[node] * dis[node];
        float4 v = ((const float4*)xt)[i];
        ((float4*)out)[i] = make_float4(w * v.x, w * v.y, w * v.z, w * v.w);
    }
}

// ---------------- K3: edge scatter with fp32 hardware atomics ----------------
__launch_bounds__(256)
__global__ void scatter(const long long* __restrict__ ei, const float* __restrict__ xt,
                        const float* __restrict__ dis, float* __restrict__ out, int nEdges) {
    const int lane = threadIdx.x & 31;
    const int e = blockIdx.x * 8 + (threadIdx.x >> 5);
    if (e >= nEdges) return;
    const int r = (int)ei[e];            // destination
    const int c = (int)ei[nEdges + e];   // source
    const float w = dis[r] * dis[c];
    const float4 v = ((const float4*)(xt + (size_t)c * DD))[lane];
    float* o = out + (size_t)r * DD + lane * 4;
    __hip_atomic_fetch_add(o + 0, w * v.x, __ATOMIC_RELAXED, __HIP_MEMORY_SCOPE_AGENT);
    __hip_atomic_fetch_add(o + 1, w * v.y, __ATOMIC_RELAXED, __HIP_MEMORY_SCOPE_AGENT);
    __hip_atomic_fetch_add(o + 2, w * v.z, __ATOMIC_RELAXED, __HIP_MEMORY_SCOPE_AGENT);
    __hip_atomic_fetch_add(o + 3, w * v.w, __ATOMIC_RELAXED, __HIP_MEMORY_SCOPE_AGENT);
}

// ---------------- K4: proj∘expmap0, leaky_relu∘logmap0, proj∘expmap0 (in place) -------
__launch_bounds__(256)
__global__ void finalize(float* __restrict__ out, int nNodes) {
    const int lane = threadIdx.x & 31;
    const int node = blockIdx.x * 8 + (threadIdx.x >> 5);
    if (node >= nNodes) return;
    float* row = out + (size_t)node * DD;
    float4 u = ((const float4*)row)[lane];

    float s = u.x * u.x + u.y * u.y + u.z * u.z + u.w * u.w;
    s = red32(s);
    float n  = fmaxf(sqrtf(s), MIN_NORM);
    float f  = tanhf(n) / n;                      // expmap0
    float hn = fmaxf(f * sqrtf(s), MIN_NORM);
    if (hn > MAXNORM) { f *= MAXNORM / hn; hn = MAXNORM; }   // proj
    float fg = f * (artanh_clip(hn) / hn);        // * logmap0

    float4 xt = make_float4(fg * u.x, fg * u.y, fg * u.z, fg * u.w);
    xt.x = xt.x > 0.0f ? xt.x : 0.01f * xt.x;
    xt.y = xt.y > 0.0f ? xt.y : 0.01f * xt.y;
    xt.z = xt.z > 0.0f ? xt.z : 0.01f * xt.z;
    xt.w = xt.w > 0.0f ? xt.w : 0.01f * xt.w;

    float s2 = xt.x * xt.x + xt.y * xt.y + xt.z * xt.z + xt.w * xt.w;
    s2 = red32(s2);
    float n2  = fmaxf(sqrtf(s2), MIN_NORM);
    float f2  = tanhf(n2) / n2;                   // expmap0
    float hn2 = fmaxf(f2 * sqrtf(s2), MIN_NORM);
    if (hn2 > MAXNORM) f2 *= MAXNORM / hn2;       // proj

    ((float4*)row)[lane] = make_float4(f2 * xt.x, f2 * xt.y, f2 * xt.z, f2 * xt.w);
}

extern "C" void kernel_launch(void* const* d_in, const int* in_sizes, int n_in,
                              void* d_out, int out_size, void* d_ws, size_t ws_size,
                              hipStream_t stream) {
    const float*      x    = (const float*)d_in[0];
    const float*      w    = (const float*)d_in[1];
    const float*      bias = (const float*)d_in[2];
    const long long*  ei   = (const long long*)d_in[3];
    float*            out  = (float*)d_out;

    const int N = in_sizes[0] / DD;   // 50000
    const int E = in_sizes[3] / 2;    // 800000

    // workspace layout (bytes): hb[0,1KB) | deg[1KB,+4N) | dis | x_t
    char* ws = (char*)d_ws;
    float*    hb  = (float*)ws;
    unsigned* deg = (unsigned*)(ws + 1024);
    float*    dis = (float*)(ws + 1024 + 200704);
    float*    xt  = (float*)(ws + 1024 + 2 * 200704);

    hb_kernel<<<1, 32, 0, stream>>>(bias, hb);
    zero_u32<<<(N + 255) / 256, 256, 0, stream>>>(deg, N);
    deg_count<<<(E + 255) / 256, 256, 0, stream>>>(ei, deg, E);
    dis_kernel<<<(N + 255) / 256, 256, 0, stream>>>(deg, dis, N);
    gemm_hyp<<<(N + 127) / 128, 256, 0, stream>>>(x, w, hb, xt, N);
    agg_init<<<(N * 32 + 255) / 256, 256, 0, stream>>>(xt, dis, out, N);
    scatter<<<(E + 7) / 8, 256, 0, stream>>>(ei, xt, dis, out, E);
    finalize<<<(N + 7) / 8, 256, 0, stream>>>(out, N);
}